// Net_42451456753895
// MI455X (gfx1250) — compile-verified
//
#include <hip/hip_runtime.h>
#include <hip/hip_bf16.h>
#include <math.h>

#define BATCH 4096
#define SEQ   512
#define NT    9
#define TDIM  11
#define SB9   (SEQ*NT)

#define WAVES_PER_BLOCK 4
#define SEQ_PER_WAVE    16

#define NEG_BIG 1.0e30f   // finite stand-in for -inf (keeps blends NaN-free)
#define LN2F    0.69314718055994530942f
#define SWZ_SWAP16 0x401F // ds_swizzle group-of-32: and=0x1f, or=0, xor=0x10 (SWAPX16)

typedef __attribute__((ext_vector_type(2))) float v2f;
typedef __attribute__((ext_vector_type(8))) float v8f;

// raw v_log_f32 (log2) scaled to natural log; inputs here are >= 1e-37, no guards needed
__device__ __forceinline__ float fast_log(float x){
  return LN2F * __builtin_amdgcn_logf(x);
}
// exchange with the other 16-lane half (lane ^ 16) via immediate-pattern swizzle
__device__ __forceinline__ float swap16(float x){
  return __int_as_float(__builtin_amdgcn_ds_swizzle(__float_as_int(x), SWZ_SWAP16));
}

// Forward algorithm, transposed state: D^T = E^T @ P^T per step.
//   A operand (constant) = E^T: A[j][i] = exp(T[i][j]), zero-padded (j,i >= 9 -> 0)
//   B operand = P^T:       B[i][m] = exp(alpha[m][i] - c_m)
//   C/D layout => lane owns sequence m = lane&15; tags j = v + 8*half live in alpha[v].
// c_m = alpha[m][tag0]: a valid normalizer (within-sequence spread is bounded by the
// em spread + 0.2, far below f32 exp range). Dead tags ride at finite huge-negative
// values; their P entries multiply the zero rows of E, never touching live results.
__global__ __launch_bounds__(WAVES_PER_BLOCK*32)
void crf_forward_kernel(const float* __restrict__ bert,
                        const int*   __restrict__ mask,
                        const float* __restrict__ T,
                        float*       __restrict__ totals)
{
  const int lane = threadIdx.x & 31;
  const int wid  = threadIdx.x >> 5;
  const int b0   = (blockIdx.x * WAVES_PER_BLOCK + wid) * SEQ_PER_WAVE;
  const int m    = lane & 15;   // sequence owned by this lane (also A-row j for bm)
  const int half = lane >> 4;   // 0: tags 0-7, 1: tags 8-15

  // A operand registers: chunk c slot k -> E^T[lane&15][4c+2*half+k] = E[r][m]
  v2f bm[3];
  #pragma unroll
  for (int c = 0; c < 3; ++c){
    #pragma unroll
    for (int k = 0; k < 2; ++k){
      int r  = 4*c + 2*half + k;               // K index (source tag i)
      int rc = (r < NT) ? r : (NT-1);
      int jc = (m < NT) ? m : (NT-1);
      float e = __expf(T[rc*TDIM + jc]);
      bm[c][k] = (r < NT && m < NT) ? e : 0.0f;
    }
  }

  // h=1 reads a +1-shifted 8-float window so its live tag 8 lands in slot 7;
  // offsets stay lane-uniform immediates and the last element is in-bounds.
  const float* bp = bert + (b0 + m)*SB9 + half;
  const int*   mp = mask + (b0 + m)*SEQ;

  // alpha0[j] = T[start=9][j] + em[:,0,:]; dead tags -> -1e30 (finite)
  float alpha[8];
  {
    float L[8];
    #pragma unroll
    for (int v = 0; v < 8; ++v) L[v] = bp[v];
    #pragma unroll
    for (int v = 0; v < 8; ++v){
      float em = half ? ((v == 0) ? L[7] : 0.0f) : L[v];
      float ts = half ? T[NT*TDIM + NT - 1] : T[NT*TDIM + v];
      bool valid = (half == 0) || (v == 0);
      alpha[v] = valid ? (ts + em) : -NEG_BIG;
    }
  }

  for (int s = 1; s < SEQ; ++s){
    // all global traffic for this step: 8 contiguous em floats + 1 mask word
    float L[8];
    #pragma unroll
    for (int v = 0; v < 8; ++v) L[v] = bp[s*NT + v];
    float mf = (float)mp[s];

    // one swizzle round: partner-half tags needed for the B operand + normalizer
    float sa0 = swap16(alpha[0]);
    float sa1 = swap16(alpha[1]);
    float sa2 = swap16(alpha[2]);
    float sa3 = swap16(alpha[3]);
    float sa6 = swap16(alpha[6]);
    float sa7 = swap16(alpha[7]);

    float c0 = half ? sa0 : alpha[0];          // tag-0 value of this sequence

    // B = P^T: chunk c slot k holds P[m][4c+2*half+k]
    v2f pb0, pb1, pb2;
    pb0[0] = __expf((half ? sa2 : alpha[0]) - c0);  // tags 0,1 | 2,3
    pb0[1] = __expf((half ? sa3 : alpha[1]) - c0);
    pb1[0] = __expf((half ? sa6 : alpha[4]) - c0);  // tags 4,5 | 6,7
    pb1[1] = __expf((half ? sa7 : alpha[5]) - c0);
    pb2[0] = __expf((half ? alpha[2] : sa0) - c0);  // tags 8,9 | 10,11
    pb2[1] = __expf((half ? alpha[3] : sa1) - c0);

    v8f d = {};
    d = __builtin_amdgcn_wmma_f32_16x16x4_f32(false, bm[0], false, pb0, (short)0, d, false, false);
    d = __builtin_amdgcn_wmma_f32_16x16x4_f32(false, bm[1], false, pb1, (short)0, d, false, false);
    d = __builtin_amdgcn_wmma_f32_16x16x4_f32(false, bm[2], false, pb2, (short)0, d, false, false);

    #pragma unroll
    for (int v = 0; v < 8; ++v){
      float em = half ? ((v == 0) ? L[7] : 0.0f) : L[v];
      float dl = fmaxf(d[v], 1.0e-37f);        // live: >=0.9; dead: clamp for finite log
      float nv = em + c0 + fast_log(dl);
      alpha[v] = alpha[v] + mf * (nv - alpha[v]); // branchless masked update
    }
  }

  // total_m = logsumexp_j(alpha[j] + T[j][end=10]) : 8 register-local + partner half
  float t[8];
  #pragma unroll
  for (int v = 0; v < 8; ++v){
    float te = half ? ((v == 0) ? T[(NT-1)*TDIM + NT + 1] : -NEG_BIG)
                    : T[v*TDIM + NT + 1];
    t[v] = alpha[v] + te;
  }
  float mxl = t[0];
  #pragma unroll
  for (int v = 1; v < 8; ++v) mxl = fmaxf(mxl, t[v]);
  float mx = fmaxf(mxl, swap16(mxl));
  float sml = 0.0f;
  #pragma unroll
  for (int v = 0; v < 8; ++v) sml += __expf(t[v] - mx);  // dead: exp(-huge) = 0
  float sm = sml + swap16(sml);
  if (half == 0) totals[b0 + m] = mx + fast_log(sm);     // coalesced, lanes 0-15
}

// Gold-path score per sequence (branchless): emit + first + mid + last, plus length.
__global__ void crf_real_kernel(const float* __restrict__ bert,
                                const int*   __restrict__ mask,
                                const int*   __restrict__ tags,
                                const float* __restrict__ T,
                                float* __restrict__ reals,
                                float* __restrict__ lens)
{
  int b = blockIdx.x * blockDim.x + threadIdx.x;
  if (b >= BATCH) return;
  const int*   mrow = mask + b*SEQ;
  const int*   trow = tags + b*SEQ;
  const float* brow = bert + b*SB9;
  float emit = 0.0f, mid = 0.0f;
  int len = 0, prevt = 0, lastt = 0;
  for (int s = 0; s < SEQ; ++s){
    int mk = mrow[s];
    int t  = trow[s];
    float mf = (float)mk;
    emit += mf * brow[s*NT + t];                 // gather always in-bounds (t<9)
    if (s > 0) mid += mf * T[prevt*TDIM + t];
    lastt = mk ? t : lastt;
    len  += mk;
    prevt = t;
  }
  float first = T[NT*TDIM + trow[0]];
  float last  = T[lastt*TDIM + NT + 1];
  reals[b] = emit + first + mid + last;
  lens[b]  = (float)len;
}

// Final scalar: sum(total - real) / sum(len)
__global__ void crf_reduce_kernel(const float* __restrict__ totals,
                                  const float* __restrict__ reals,
                                  const float* __restrict__ lens,
                                  float* __restrict__ out)
{
  __shared__ float sh1[8], sh2[8];
  float s1 = 0.0f, s2 = 0.0f;
  for (int i = threadIdx.x; i < BATCH; i += blockDim.x){
    s1 += totals[i] - reals[i];
    s2 += lens[i];
  }
  for (int off = 16; off >= 1; off >>= 1){
    s1 += __shfl_xor(s1, off, 32);
    s2 += __shfl_xor(s2, off, 32);
  }
  int lane = threadIdx.x & 31, wid = threadIdx.x >> 5;
  if (lane == 0){ sh1[wid] = s1; sh2[wid] = s2; }
  __syncthreads();
  if (threadIdx.x == 0){
    float t1 = 0.0f, t2 = 0.0f;
    for (int w = 0; w < (int)(blockDim.x >> 5); ++w){ t1 += sh1[w]; t2 += sh2[w]; }
    out[0] = t1 / t2;
  }
}

extern "C" void kernel_launch(void* const* d_in, const int* in_sizes, int n_in,
                              void* d_out, int out_size, void* d_ws, size_t ws_size,
                              hipStream_t stream)
{
  const float* bert = (const float*)d_in[0];   // (4096, 512, 9) f32
  const int*   mask = (const int*)d_in[1];     // (4096, 512) i32
  const int*   tags = (const int*)d_in[2];     // (4096, 512) i32 (jax downgrades int64)
  const float* T    = (const float*)d_in[3];   // (11, 11) f32
  float* out    = (float*)d_out;
  float* totals = (float*)d_ws;                // [4096]
  float* reals  = totals + BATCH;              // [4096]
  float* lens   = reals  + BATCH;              // [4096]

  crf_real_kernel<<<BATCH/256, 256, 0, stream>>>(bert, mask, tags, T, reals, lens);

  const int nblocks = BATCH / (WAVES_PER_BLOCK * SEQ_PER_WAVE); // 64 blocks x 4 waves
  crf_forward_kernel<<<nblocks, WAVES_PER_BLOCK*32, 0, stream>>>(bert, mask, T, totals);

  crf_reduce_kernel<<<1, 256, 0, stream>>>(totals, reals, lens, out);
}